// MarketImpactHedger_48473000902862
// MI455X (gfx1250) — compile-verified
//
#include <hip/hip_runtime.h>

// MarketImpactHedger fused kernel for MI455X (gfx1250, wave32).
//
// One wave owns 32 batch rows, loops t=0..255:
//   base MLP 3->64(relu)->1 : V_WMMA_F32_16X16X4_F32, K=4 = 3 features + bias-1
//   relu                    : v_med3_num_f32(c, 0, FLT_MAX)  (1 instr, no canon)
//   layer-2 dot (h . w2)    : v2f pairs over adjacent C regs -> v_pk_fma_f32
//   row reduction           : reg/lane merge-butterfly on v_permlane16_b32
//   scan + 3->16->1 tanh net: v2f-packed hidden layer (v_pk_fma_f32), v_tanh_f32
// Feature loads are software-pipelined one 4-step block ahead (b128 clauses).

#define BB 65536
#define TT 256

// Flip to 0 if gfx1250 backend cannot select v_permlane16/x16 (it can).
#ifndef USE_PERMLANE
#define USE_PERMLANE 1
#endif
#if USE_PERMLANE && __has_builtin(__builtin_amdgcn_permlane16) && \
    __has_builtin(__builtin_amdgcn_permlanex16)
#define HAVE_PERMLANE 1
#else
#define HAVE_PERMLANE 0
#endif

typedef float v2f __attribute__((ext_vector_type(2)));
typedef float v8f __attribute__((ext_vector_type(8)));

#define RELU_HI 3.402823466e38f

// ds_swizzle xor-mask fallback (group-of-32 mode: and=0x1f, or=0, xor=X)
template <int XORM>
__device__ __forceinline__ float swz(float v) {
  int r = __builtin_amdgcn_ds_swizzle(__builtin_bit_cast(int, v),
                                      (XORM << 10) | 0x1f);
  return __builtin_bit_cast(float, r);
}

// permlane selector word: nibble at position i (+base) = (i ^ d) & 15
constexpr unsigned selword(int d, int base) {
  unsigned w = 0;
  for (int i = 0; i < 8; ++i)
    w |= (unsigned)(((base + i) ^ d) & 15) << (4 * i);
  return w;
}

// xor-shuffle WITHIN each 16-lane half (D in {1,2,4,8})
template <int D>
__device__ __forceinline__ float xsh_in16(float v) {
#if HAVE_PERMLANE
  int s = __builtin_bit_cast(int, v);
  int r = __builtin_amdgcn_permlane16(s, s, selword(D, 0), selword(D, 8),
                                      false, false);
  return __builtin_bit_cast(float, r);
#else
  return swz<D>(v);
#endif
}

// cross-half shuffle: lane p <-> other half at position p^D (D in {0,8})
template <int D>
__device__ __forceinline__ float xsh_x16(float v) {
#if HAVE_PERMLANE
  int s = __builtin_bit_cast(int, v);
  int r = __builtin_amdgcn_permlanex16(s, s, selword(D, 0), selword(D, 8),
                                       false, false);
  return __builtin_bit_cast(float, r);
#else
  return swz<(16 | D)>(v);
#endif
}

// Merge two row-registers a(row bit=0), b(row bit=1) keyed on lane bit D:
// result lane l: row selected by (l & D), partial-summed over xor distance D.
template <int D>
__device__ __forceinline__ float mrg(float a, float b, int lane) {
  float sa = xsh_in16<D>(a);
  float sb = xsh_in16<D>(b);
  bool hi = (lane & D) != 0;
  return (hi ? b : a) + (hi ? sb : sa);
}

__device__ __forceinline__ float fast_tanh(float x) {
#if __has_builtin(__builtin_amdgcn_tanhf)
  return __builtin_amdgcn_tanhf(x);   // v_tanh_f32 (CDNA5 trans op)
#else
  float ax = __builtin_fabsf(x);
  float e  = __expf(-2.0f * ax);
  float r  = 1.0f - 2.0f * (e / (1.0f + e));
  return x < 0.0f ? -r : r;
#endif
}

__global__ __launch_bounds__(256)
void hedger_fused_kernel(const float* __restrict__ features,
                         const float* __restrict__ w1,  // (3,64)
                         const float* __restrict__ b1,  // (64)
                         const float* __restrict__ w2,  // (64,1)
                         const float* __restrict__ b2,  // (1)
                         const float* __restrict__ aw1, // (3,16)
                         const float* __restrict__ ab1, // (16)
                         const float* __restrict__ aw2, // (16,1)
                         const float* __restrict__ ab2, // (1)
                         float* __restrict__ out)       // (B,T)
{
  const int lane = threadIdx.x & 31;
  const int wave = blockIdx.x * (blockDim.x >> 5) + (threadIdx.x >> 5);
  const int b0   = wave * 32;          // 32 batch rows per wave
  const int row  = b0 + lane;          // this lane's batch row
  const bool lo  = lane < 16;
  const int nlo  = lane & 15;

  // ---- B matrices: w1 with base_b1 folded in as the K=3 row (bias) ----
  // B 4x16 f32 layout: VGPR0 = {K0 | K2}, VGPR1 = {K1 | K3} across lane halves.
  v2f   Bm[4];
  float w2v[4];
#pragma unroll
  for (int nt = 0; nt < 4; ++nt) {
    int n = nt * 16 + nlo;
    Bm[nt].x = lo ? w1[0 * 64 + n] : w1[2 * 64 + n]; // K0 | K2
    Bm[nt].y = lo ? w1[1 * 64 + n] : b1[n];          // K1 | K3(bias row)
    w2v[nt]  = w2[n];                                // col weight per lane
  }
  const float bb2 = b2[0];

  // ---- adjustment-net weights: wave-uniform, packed into v2f pairs ----
  v2f A1wv[3][8], A1bv[8], A2wv[8];
#pragma unroll
  for (int k = 0; k < 3; ++k)
#pragma unroll
    for (int j = 0; j < 8; ++j) {
      A1wv[k][j].x = aw1[k * 16 + 2 * j];
      A1wv[k][j].y = aw1[k * 16 + 2 * j + 1];
    }
#pragma unroll
  for (int j = 0; j < 8; ++j) {
    A1bv[j].x = ab1[2 * j];     A1bv[j].y = ab1[2 * j + 1];
    A2wv[j].x = aw2[2 * j];     A2wv[j].y = aw2[2 * j + 1];
  }
  const float ab2v = ab2[0];

  const float* fb = features + (size_t)row * (TT * 3);
  float*       ob = out + (size_t)row * TT;

  float prev = 0.0f;
  const v8f zero8 = {0.f, 0.f, 0.f, 0.f, 0.f, 0.f, 0.f, 0.f};

  // software pipeline: features for the current 4-step block live in cx*
  float cx0[4], cx1[4], cx2[4];
#pragma unroll
  for (int ti = 0; ti < 4; ++ti) {
    cx0[ti] = fb[3 * ti + 0];
    cx1[ti] = fb[3 * ti + 1];
    cx2[ti] = fb[3 * ti + 2];
  }

  for (int t4 = 0; t4 < TT; t4 += 4) {
    // prefetch next block (branchless wrap at the end; extra loads harmless)
    const int tn = (t4 + 4) & (TT - 1);
    float nx0[4], nx1[4], nx2[4];
#pragma unroll
    for (int ti = 0; ti < 4; ++ti) {
      nx0[ti] = fb[3 * (tn + ti) + 0];
      nx1[ti] = fb[3 * (tn + ti) + 1];
      nx2[ti] = fb[3 * (tn + ti) + 2];
    }

    float o4[4];
#pragma unroll
    for (int ti = 0; ti < 4; ++ti) {
      float x0 = cx0[ti], x1 = cx1[ti], x2 = cx2[ti];

      // ---- A tiles (16x4 f32: lanes 0-15 K{0,1}, lanes 16-31 K{2,3}) ----
      float sx0 = xsh_x16<0>(x0), sx1 = xsh_x16<0>(x1), sx2 = xsh_x16<0>(x2);
      v2f A0, A1m;
      A0.x  = lo ? x0  : sx2;   // tile0 = rows b0..b0+15
      A0.y  = lo ? x1  : 1.0f;  // K3 = 1 -> bias
      A1m.x = lo ? sx0 : x2;    // tile1 = rows b0+16..b0+31
      A1m.y = lo ? sx1 : 1.0f;

      // ---- base MLP: 8x WMMA f32 16x16x4; relu(med3) + pk_fma dot ----
      v2f Pv0[4], Pv1[4];
#pragma unroll
      for (int k = 0; k < 4; ++k) {
        Pv0[k].x = 0.f; Pv0[k].y = 0.f;
        Pv1[k].x = 0.f; Pv1[k].y = 0.f;
      }
#pragma unroll
      for (int nt = 0; nt < 4; ++nt) {
        v8f c0 = __builtin_amdgcn_wmma_f32_16x16x4_f32(
            false, A0, false, Bm[nt], (short)0, zero8, false, false);
        v8f c1 = __builtin_amdgcn_wmma_f32_16x16x4_f32(
            false, A1m, false, Bm[nt], (short)0, zero8, false, false);
        v2f wp; wp.x = w2v[nt]; wp.y = w2v[nt];
#pragma unroll
        for (int k = 0; k < 4; ++k) {
          v2f h0, h1;   // adjacent C elements -> adjacent VGPRs, no movs
          h0.x = __builtin_amdgcn_fmed3f(c0[2 * k + 0], 0.f, RELU_HI);
          h0.y = __builtin_amdgcn_fmed3f(c0[2 * k + 1], 0.f, RELU_HI);
          h1.x = __builtin_amdgcn_fmed3f(c1[2 * k + 0], 0.f, RELU_HI);
          h1.y = __builtin_amdgcn_fmed3f(c1[2 * k + 1], 0.f, RELU_HI);
          Pv0[k] = h0 * wp + Pv0[k];   // v_pk_fma_f32
          Pv1[k] = h1 * wp + Pv1[k];
        }
      }
      float P0[8], P1[8];
#pragma unroll
      for (int k = 0; k < 4; ++k) {    // pure register aliasing
        P0[2 * k] = Pv0[k].x; P0[2 * k + 1] = Pv0[k].y;
        P1[2 * k] = Pv1[k].x; P1[2 * k + 1] = Pv1[k].y;
      }

      // ---- merge-butterfly: 8 regs x 16-lane halves -> lane l = row sum ----
      float q0, q1, q2, q3, r0, r1, s0, s1;
      q0 = mrg<4>(P0[0], P0[4], lane); q1 = mrg<4>(P0[1], P0[5], lane);
      q2 = mrg<4>(P0[2], P0[6], lane); q3 = mrg<4>(P0[3], P0[7], lane);
      r0 = mrg<2>(q0, q2, lane);       r1 = mrg<2>(q1, q3, lane);
      s0 = mrg<1>(r0, r1, lane);       s0 += xsh_in16<8>(s0);
      q0 = mrg<4>(P1[0], P1[4], lane); q1 = mrg<4>(P1[1], P1[5], lane);
      q2 = mrg<4>(P1[2], P1[6], lane); q3 = mrg<4>(P1[3], P1[7], lane);
      r0 = mrg<2>(q0, q2, lane);       r1 = mrg<2>(q1, q3, lane);
      s1 = mrg<1>(r0, r1, lane);       s1 += xsh_in16<8>(s1);
      // route: lane l gets raw[b0+l]
      float raw0 = ((lane & 24) == 8) ? xsh_x16<8>(s0) : s0;        // rows 0..15
      float raw1 = (lane < 24) ? xsh_x16<0>(s1) : xsh_in16<8>(s1);  // rows 16..31
      float raw  = (lo ? raw0 : raw1) + bb2;

      // ---- sequential scan step (all 32 lanes, one row each) ----
      float dc = raw - prev;
      v2f rawv; rawv.x = raw; rawv.y = raw;
      v2f dcv;  dcv.x  = dc;  dcv.y  = dc;
      v2f x2v;  x2v.x  = x2;  x2v.y  = x2;
      v2f accv; accv.x = ab2v; accv.y = 0.f;
#pragma unroll
      for (int j = 0; j < 8; ++j) {     // 2 hidden units per iteration
        v2f h = rawv * A1wv[0][j] +
                (dcv * A1wv[1][j] + (x2v * A1wv[2][j] + A1bv[j]));
        h.x = __builtin_amdgcn_fmed3f(h.x, 0.f, RELU_HI);
        h.y = __builtin_amdgcn_fmed3f(h.y, 0.f, RELU_HI);
        accv = h * A2wv[j] + accv;      // v_pk_fma_f32
      }
      float acc = accv.x + accv.y;

      float adj     = fast_tanh(acc);
      float clipped = __builtin_fminf(__builtin_fmaxf(dc, -0.5f), 0.5f);
      float fac     = __builtin_fmaf(-0.1f, adj, 1.0f);
      float nd      = __builtin_fmaf(clipped, fac, prev);
      nd   = __builtin_fminf(__builtin_fmaxf(nd, -1.5f), 1.5f);
      prev = nd;
      o4[ti] = nd;
    }
    float4 v; v.x = o4[0]; v.y = o4[1]; v.z = o4[2]; v.w = o4[3];
    *reinterpret_cast<float4*>(ob + t4) = v;   // 16B-aligned b128 store

#pragma unroll
    for (int ti = 0; ti < 4; ++ti) {
      cx0[ti] = nx0[ti]; cx1[ti] = nx1[ti]; cx2[ti] = nx2[ti];
    }
  }
}

extern "C" void kernel_launch(void* const* d_in, const int* in_sizes, int n_in,
                              void* d_out, int out_size, void* d_ws,
                              size_t ws_size, hipStream_t stream) {
  const float* features = (const float*)d_in[0];
  const float* w1  = (const float*)d_in[1];
  const float* b1  = (const float*)d_in[2];
  const float* w2  = (const float*)d_in[3];
  const float* b2  = (const float*)d_in[4];
  const float* aw1 = (const float*)d_in[5];
  const float* ab1 = (const float*)d_in[6];
  const float* aw2 = (const float*)d_in[7];
  const float* ab2 = (const float*)d_in[8];
  float* out = (float*)d_out;

  // 32 rows/wave * 8 waves/block = 256 rows per block -> 256 blocks
  dim3 grid(BB / 256), block(256);
  hipLaunchKernelGGL(hedger_fused_kernel, grid, block, 0, stream,
                     features, w1, b1, w2, b2, aw1, ab1, aw2, ab2, out);
}